// Actions_Emb_40261023432786
// MI455X (gfx1250) — compile-verified
//
#include <hip/hip_runtime.h>

typedef __attribute__((ext_vector_type(2))) float v2f;
typedef __attribute__((ext_vector_type(8))) float v8f;

#define BDIM    128      // 4 wave32
#define SLOTS   16       // (b,s) slots per block -> A-matrix M dimension
#define KPAD    64       // histogram width padded (N_CHAR=58)
#define KSTEPS  15       // K=60 >= 58, 4 per wmma step
#define NCHAR   58
#define DMODEL  256
#define BOS_ROW 98

__global__ __launch_bounds__(BDIM)
void actions_emb_wmma_kernel(const int* __restrict__ char_ids,      // [B,S,16]
                             const int* __restrict__ char_len,      // [B,S]
                             const int* __restrict__ action_ids,    // [B,S]
                             const int* __restrict__ slot_type,     // [B,S]
                             const float* __restrict__ char_table,  // [58,256]
                             const float* __restrict__ action_table,// [99,256]
                             float* __restrict__ out)               // [B,5,256]
{
    __shared__ float H[SLOTS][KPAD];   // A-matrix: scaled histogram (4 KB)
    __shared__ int   sT[SLOTS];
    __shared__ int   sAid[SLOTS];

    const int tid   = threadIdx.x;
    const int g     = blockIdx.x;      // block of 16 flat slots
    const int slot0 = g * SLOTS;       // flat slot = b*4 + s

    // ---- Phase 0: zero histogram ----
    for (int i = tid; i < SLOTS * KPAD; i += BDIM) (&H[0][0])[i] = 0.0f;
    __syncthreads();

    // ---- Phase 1: build scaled histogram rows (one thread per slot, no atomics) ----
    if (tid < SLOTS) {
        const int gs = slot0 + tid;
        const int t  = slot_type[gs];
        sT[tid]   = t;
        sAid[tid] = action_ids[gs];
        if (t == 0) {
            const int   len = char_len[gs];
            const float inv = 1.0f / (float)len;
            const int* ids  = char_ids + gs * 16;
            for (int l = 0; l < len; ++l) {
                H[tid][ids[l]] += inv;     // row is exclusive to this thread
            }
        }
        // t != 0 rows stay zero -> WMMA yields zero (t==2 final value; t==1 overwritten)
    }
    __syncthreads();

    const int wave    = tid >> 5;
    const int lane    = tid & 31;
    const int halfSel = lane >> 4;     // 0: lanes 0-15, 1: lanes 16-31
    const int l16     = lane & 15;

    // ---- Preload A fragments (16x4 f32 layout: VGPR0 = K+0/K+2, VGPR1 = K+1/K+3) ----
    v2f afrag[KSTEPS];
    #pragma unroll
    for (int k = 0; k < KSTEPS; ++k) {
        const int kb = k * 4 + halfSel * 2;
        afrag[k].x = H[l16][kb + 0];
        afrag[k].y = H[l16][kb + 1];
    }

    // ---- Phase 2: GEMM. Each wave owns 4 of 16 column tiles of D=256 ----
    for (int nt = 0; nt < 4; ++nt) {
        const int col = (wave * 4 + nt) * 16 + l16;
        v8f acc = {};
        #pragma unroll
        for (int k = 0; k < KSTEPS; ++k) {
            const int r0 = k * 4 + halfSel * 2;   // B 4x16: same half-wave striping as A
            v2f bfrag;
            bfrag.x = (r0 + 0 < NCHAR) ? char_table[(r0 + 0) * DMODEL + col] : 0.0f;
            bfrag.y = (r0 + 1 < NCHAR) ? char_table[(r0 + 1) * DMODEL + col] : 0.0f;
            acc = __builtin_amdgcn_wmma_f32_16x16x4_f32(
                false, afrag[k], false, bfrag, (short)0, acc, false, false);
        }

        // ---- Phase 3: slot_type select + store. acc[i] = row (i + halfSel*8), this col ----
        #pragma unroll
        for (int i = 0; i < 8; ++i) {
            const int m  = i + halfSel * 8;       // slot within block
            float val = acc[i];
            if (sT[m] == 1) val = action_table[sAid[m] * DMODEL + col];
            const int gs = slot0 + m;
            const int b  = gs >> 2;               // S == 4
            const int s  = gs & 3;
            out[((long)b * 5 + (s + 1)) * DMODEL + col] = val;
        }
    }

    // ---- BOS rows: this block owns b = 4g .. 4g+3; out[b][0][:] = action_table[98] ----
    for (int i = tid; i < 4 * DMODEL; i += BDIM) {
        const int bb = i >> 8;
        const int d  = i & 255;
        out[(long)(4 * g + bb) * 5 * DMODEL + d] = action_table[BOS_ROW * DMODEL + d];
    }
}

extern "C" void kernel_launch(void* const* d_in, const int* in_sizes, int n_in,
                              void* d_out, int out_size, void* d_ws, size_t ws_size,
                              hipStream_t stream) {
    const int*   char_ids     = (const int*)d_in[0];
    const int*   char_len     = (const int*)d_in[1];
    const int*   action_ids   = (const int*)d_in[2];
    const int*   slot_type    = (const int*)d_in[3];
    const float* char_table   = (const float*)d_in[4];
    const float* action_table = (const float*)d_in[5];
    float*       out          = (float*)d_out;

    const int total_slots = in_sizes[1];       // B*S = 65536
    const int blocks      = total_slots / SLOTS;

    hipLaunchKernelGGL(actions_emb_wmma_kernel, dim3(blocks), dim3(BDIM), 0, stream,
                       char_ids, char_len, action_ids, slot_type,
                       char_table, action_table, out);
}